// ConsistencyLoss_85899345920626
// MI455X (gfx1250) — compile-verified
//
#include <hip/hip_runtime.h>
#include <hip/hip_bf16.h>

// ConsistencyLoss for MI455X (gfx1250, wave32).
// Scatter/atomic + streaming-reduction workload; z-buffers (47MB) fit in 192MB L2.
// WMMA f32 16x16x4 used as a layout-agnostic wave/block sum reducer.

#define H_   768
#define W_   1024
#define HW_  (H_ * W_)
#define NP_  15          // number of frame pairs
#define NW_  8           // waves per 256-thread block

typedef __attribute__((ext_vector_type(2))) float v2f;
typedef __attribute__((ext_vector_type(8))) float v8f;

__device__ __forceinline__ float pos_inf() { return __uint_as_float(0x7f800000u); }

// Layout-agnostic full-wave (32-lane) sum using V_WMMA_F32_16X16X4_F32.
// A holds (x, 0) per lane (any bijective packing of the 64 A slots works since
// B is all-ones and we sum every D slot): sum over all 256 D elements = 16 * total.
__device__ __forceinline__ float wmma_wave_sum(float x) {
  v2f a; a[0] = x;   a[1] = 0.0f;
  v2f b; b[0] = 1.0f; b[1] = 1.0f;
  v8f c = {};
  c = __builtin_amdgcn_wmma_f32_16x16x4_f32(
      /*neg_a=*/false, a, /*neg_b=*/false, b,
      /*c_mod=*/(short)0, c, /*reuse_a=*/false, /*reuse_b=*/false);
  float s = ((c[0] + c[1]) + (c[2] + c[3])) + ((c[4] + c[5]) + (c[6] + c[7]));
  #pragma unroll
  for (int m = 16; m >= 1; m >>= 1) s += __shfl_xor(s, m, 32);
  return s * 0.0625f;   // /16 is exact
}

// ---------------------------------------------------------------- setup -----
__global__ void setup_kernel(const float* __restrict__ pose, float* __restrict__ xf) {
  int p = threadIdx.x;
  if (p >= NP_) return;
  const float* pA = pose + p * 7;
  const float* pB = pose + (p + 1) * 7;
  float* X = xf + p * 24;
  // quat_to_rot (normalizes like the reference)
  #pragma unroll
  for (int s = 0; s < 2; ++s) {
    const float* q = (s == 0 ? pA : pB) + 3;
    float n = sqrtf(q[0]*q[0] + q[1]*q[1] + q[2]*q[2] + q[3]*q[3]);
    float x = q[0]/n, y = q[1]/n, z = q[2]/n, w = q[3]/n;
    float* R = X + s * 9;
    R[0] = 1.f - 2.f*(y*y + z*z); R[1] = 2.f*(x*y - z*w);       R[2] = 2.f*(x*z + y*w);
    R[3] = 2.f*(x*y + z*w);       R[4] = 1.f - 2.f*(x*x + z*z); R[5] = 2.f*(y*z - x*w);
    R[6] = 2.f*(x*z - y*w);       R[7] = 2.f*(y*z + x*w);       R[8] = 1.f - 2.f*(x*x + y*y);
  }
  X[18] = pA[0]; X[19] = pA[1]; X[20] = pA[2];   // tA
  X[21] = pB[0]; X[22] = pB[1]; X[23] = pB[2];   // tB
}

// ----------------------------------------------------------------- init -----
__global__ __launch_bounds__(256) void init_kernel(float* __restrict__ buf,
                                                   float* __restrict__ pairSum,
                                                   float* __restrict__ pairCnt) {
  long i = (long)blockIdx.x * blockDim.x + threadIdx.x;
  if (i < NP_) { pairSum[i] = 0.0f; pairCnt[i] = 0.0f; }
  const float inf = pos_inf();
  float4 v = make_float4(inf, inf, inf, inf);
  const long n4 = (long)NP_ * HW_ / 4;
  const long stride = (long)gridDim.x * blockDim.x;
  for (long j = i; j < n4; j += stride) ((float4*)buf)[j] = v;
}

// --------------------------------------------------------------- scatter ----
__global__ __launch_bounds__(256) void scatter_kernel(const float* __restrict__ pred,
                                                      const float* __restrict__ xf,
                                                      const float* __restrict__ K,
                                                      float* __restrict__ buf) {
  const int pair = blockIdx.y;
  const float fx = K[0], cx = K[2], fy = K[4], cy = K[5];
  const float* X = xf + pair * 24;   // uniform -> scalar loads
  float RA[9], RB[9], tA[3], tB[3];
  #pragma unroll
  for (int i = 0; i < 9; ++i) { RA[i] = X[i]; RB[i] = X[9 + i]; }
  #pragma unroll
  for (int i = 0; i < 3; ++i) { tA[i] = X[18 + i]; tB[i] = X[21 + i]; }

  const float* dA = pred + (size_t)pair * HW_;
  unsigned* zb = (unsigned*)(buf + (size_t)pair * HW_);

  int g = blockIdx.x * blockDim.x + threadIdx.x;   // one float4 per thread
  int base = g * 4;
  float vv = (float)(base >> 10);                  // row (W = 1024)
  int u0 = base & (W_ - 1);
  float4 d4 = ((const float4*)dA)[g];
  float dv[4] = {d4.x, d4.y, d4.z, d4.w};

  #pragma unroll
  for (int j = 0; j < 4; ++j) {
    float d = dv[j];
    if (d == 0.0f) continue;                       // validA
    float u = (float)(u0 + j);
    float x = (u - cx) * d / fx;
    float y = (vv - cy) * d / fy;
    // pw = RA @ pc + tA
    float pwx = RA[0]*x + RA[1]*y + RA[2]*d + tA[0];
    float pwy = RA[3]*x + RA[4]*y + RA[5]*d + tA[1];
    float pwz = RA[6]*x + RA[7]*y + RA[8]*d + tA[2];
    // q = RB^T @ (pw - tB)
    float ax = pwx - tB[0], ay = pwy - tB[1], az = pwz - tB[2];
    float qx = RB[0]*ax + RB[3]*ay + RB[6]*az;
    float qy = RB[1]*ax + RB[4]*ay + RB[7]*az;
    float qz = RB[2]*ax + RB[5]*ay + RB[8]*az;
    if (!(qz > 0.0f)) continue;
    float u2 = fx * qx / qz + cx;
    float v2 = fy * qy / qz + cy;
    float ur = rintf(u2), vr = rintf(v2);          // round-half-even == jnp.round
    if (ur < 0.0f || ur > (float)(W_ - 1) || vr < 0.0f || vr > (float)(H_ - 1)) continue;
    int ui = (int)ur, vi = (int)vr;
    // z > 0 and init = +inf: uint order == float order for positives
    atomicMin(&zb[vi * W_ + ui], __float_as_uint(qz));
  }
}

// ---------------------------------------------------------------- reduce ----
__global__ __launch_bounds__(256) void reduce_kernel(const float* __restrict__ pred,
                                                     const float* __restrict__ buf,
                                                     float* __restrict__ pairSum,
                                                     float* __restrict__ pairCnt) {
  const int pair = blockIdx.y;
  const float4* dB = (const float4*)(pred + (size_t)(pair + 1) * HW_);
  const float4* zb = (const float4*)(buf + (size_t)pair * HW_);
  const float inf = pos_inf();

  float s = 0.0f, c = 0.0f;
  const int n4 = HW_ / 4;
  const int stride = gridDim.x * blockDim.x;
  for (int g = blockIdx.x * blockDim.x + threadIdx.x; g < n4; g += stride) {
    float4 b4 = zb[g];
    float4 p4 = dB[g];
    float bb[4] = {b4.x, b4.y, b4.z, b4.w};
    float dd[4] = {p4.x, p4.y, p4.z, p4.w};
    #pragma unroll
    for (int j = 0; j < 4; ++j) {
      if (bb[j] < inf) { s += bb[j] - dd[j]; c += 1.0f; }  // hit: repro=buf(>0)
      else if (dd[j] != 0.0f) { c += 1.0f; }               // miss: repro=dB, diff=0
    }
  }

  __shared__ float sS[NW_], sC[NW_];
  const int lane = threadIdx.x & 31;
  const int wid  = threadIdx.x >> 5;
  float wsum = wmma_wave_sum(s);
  float wcnt = wmma_wave_sum(c);
  if (lane == 0) { sS[wid] = wsum; sC[wid] = wcnt; }
  __syncthreads();
  if (wid == 0) {
    float s2 = (lane < NW_) ? sS[lane] : 0.0f;
    float c2 = (lane < NW_) ? sC[lane] : 0.0f;
    s2 = wmma_wave_sum(s2);
    c2 = wmma_wave_sum(c2);
    if (lane == 0) {
      atomicAdd(&pairSum[pair], s2);
      atomicAdd(&pairCnt[pair], c2);
    }
  }
}

// -------------------------------------------------------------- finalize ----
__global__ void finalize_kernel(const float* __restrict__ pairSum,
                                const float* __restrict__ pairCnt,
                                float* __restrict__ out) {
  if (threadIdx.x == 0 && blockIdx.x == 0) {
    float total = 0.0f;
    for (int p = 0; p < NP_; ++p) {
      float cnt = fmaxf(pairCnt[p], 1.0f);
      total += pairSum[p] / cnt;
    }
    out[0] = total;
  }
}

// ---------------------------------------------------------------- launch ----
extern "C" void kernel_launch(void* const* d_in, const int* in_sizes, int n_in,
                              void* d_out, int out_size, void* d_ws, size_t ws_size,
                              hipStream_t stream) {
  const float* pred = (const float*)d_in[0];   // (16,1,768,1024) f32
  const float* pose = (const float*)d_in[1];   // (16,7) f32
  const float* K    = (const float*)d_in[2];   // (3,3) f32
  float* out = (float*)d_out;

  char* ws = (char*)d_ws;
  float* xf      = (float*)(ws);               // 15*24 floats
  float* pairSum = (float*)(ws + 2048);        // 15 floats
  float* pairCnt = (float*)(ws + 2048 + 128);  // 15 floats
  float* buf     = (float*)(ws + 4096);        // 15*H*W floats (~47 MB, fits L2)

  setup_kernel<<<1, 32, 0, stream>>>(pose, xf);
  init_kernel<<<2048, 256, 0, stream>>>(buf, pairSum, pairCnt);

  dim3 gScatter(HW_ / 4 / 256, NP_);           // 768 x 15, 1 float4/thread
  scatter_kernel<<<gScatter, 256, 0, stream>>>(pred, xf, K, buf);

  dim3 gReduce(192, NP_);                      // 4 float4s/thread
  reduce_kernel<<<gReduce, 256, 0, stream>>>(pred, buf, pairSum, pairCnt);

  finalize_kernel<<<1, 32, 0, stream>>>(pairSum, pairCnt, out);
}